// MultiScaleRoIAlign_15719580303979
// MI455X (gfx1250) — compile-verified
//
#include <hip/hip_runtime.h>
#include <hip/hip_bf16.h>
#include <math.h>

typedef __attribute__((ext_vector_type(2))) float v2f;
typedef __attribute__((ext_vector_type(8))) float v8f;

__device__ __forceinline__ int iclampi(int v, int lo, int hi) {
  return v < lo ? lo : (v > hi ? hi : v);
}

// Multi-scale RoI align, separable-bilinear formulation:
//   O(7x7) = A_y (7x14) * F (14x14 gathered corners) * A_x^T (14x7)
// padded to 16 and computed with V_WMMA_F32_16X16X4_F32 (K=16 as 4 chained
// K=4 steps). One block per (box, channel-quarter); each wave32 owns a
// 16-channel slab. All wave-uniform address components (including the wave's
// channel block) are readfirstlane'd so loads/stores select SADDR form.
__global__ __launch_bounds__(128) void msroi_wmma_kernel(
    const float* __restrict__ f0, const float* __restrict__ f1,
    const float* __restrict__ f2, const float* __restrict__ f3,
    const float* __restrict__ boxes0, const float* __restrict__ boxes1,
    float* __restrict__ out, int n0)
{
  __shared__ int   s_xs[16];
  __shared__ int   s_ys[16];
  __shared__ float s_wxh[8], s_wxl[8], s_wyh[8], s_wyl[8];
  __shared__ int   s_lvl, s_b, s_H, s_W;
  __shared__ float s_T[4][16][17];   // per-wave transpose scratch (pad=17)

  const int box = blockIdx.x;

  // ---- per-box setup (level select + corner indices + masked weights) ----
  if (threadIdx.x == 0) {
    const float* bp = (box < n0) ? (boxes0 + (size_t)box * 4)
                                 : (boxes1 + (size_t)(box - n0) * 4);
    float x1 = bp[0], y1 = bp[1], x2 = bp[2], y2 = bp[3];
    float sz  = sqrtf((x2 - x1) * (y2 - y1));
    int   lev = (int)floorf(4.0f + log2f(sz * (1.0f / 224.0f)) + 1e-8f);
    int   l   = iclampi(lev, 2, 5) - 2;          // 0..3 -> strides 4,8,16,32
    int   H   = 640  >> (l + 2);
    int   W   = 1024 >> (l + 2);
    float scale = 1.0f / (float)(4 << l);
    float x1s = x1 * scale, y1s = y1 * scale;
    float rw  = fmaxf(x2 * scale - x1s, 1.0f);
    float rh  = fmaxf(y2 * scale - y1s, 1.0f);
    float bw  = rw * (1.0f / 7.0f), bh = rh * (1.0f / 7.0f);
    #pragma unroll
    for (int p = 0; p < 7; ++p) {
      // X axis
      float xx = x1s + ((float)p + 0.5f) * bw;
      bool  vx = (xx > -1.0f) && (xx < (float)W);
      float xc = fmaxf(xx, 0.0f);
      int   xl = iclampi((int)floorf(xc), 0, W - 1);
      int   xh = (xl + 1 < W) ? xl + 1 : W - 1;
      float lx = (xl >= W - 1) ? 0.0f : (xc - (float)xl);
      float hx = 1.0f - lx;
      if (!vx) { hx = 0.0f; lx = 0.0f; }        // fold validity mask
      s_xs[2 * p] = xl; s_xs[2 * p + 1] = xh;
      s_wxh[p] = hx;    s_wxl[p] = lx;
      // Y axis
      float yy = y1s + ((float)p + 0.5f) * bh;
      bool  vy = (yy > -1.0f) && (yy < (float)H);
      float yc = fmaxf(yy, 0.0f);
      int   yl = iclampi((int)floorf(yc), 0, H - 1);
      int   yh = (yl + 1 < H) ? yl + 1 : H - 1;
      float ly = (yl >= H - 1) ? 0.0f : (yc - (float)yl);
      float hy = 1.0f - ly;
      if (!vy) { hy = 0.0f; ly = 0.0f; }
      s_ys[2 * p] = yl; s_ys[2 * p + 1] = yh;
      s_wyh[p] = hy;    s_wyl[p] = ly;
    }
    s_xs[14] = 0; s_xs[15] = 0;                  // K-pad rows/cols (weight 0)
    s_ys[14] = 0; s_ys[15] = 0;
    s_lvl = l; s_b = (box < n0) ? 0 : 1;
    s_H = H; s_W = W;
  }
  __syncthreads();

  const int lane = threadIdx.x & 31;
  const int wv   = threadIdx.x >> 5;             // wave in block: 0..3
  const int m    = lane & 15;                    // M (or N) index
  const int koff = (lane < 16) ? 0 : 2;          // lane-half splits K blocks

  // Promote wave/box-uniform values into SGPRs so address math scalarizes.
  // cb is wave-uniform (derived from the wave id) but the compiler's
  // divergence analysis can't prove it -> readfirstlane it explicitly.
  const int cbS = __builtin_amdgcn_readfirstlane(blockIdx.y * 4 + wv);
  const int lS  = __builtin_amdgcn_readfirstlane(s_lvl);
  const int bS  = __builtin_amdgcn_readfirstlane(s_b);
  const int HS  = __builtin_amdgcn_readfirstlane(s_H);
  const int WS  = __builtin_amdgcn_readfirstlane(s_W);

  const size_t planeHW    = (size_t)HS * (size_t)WS;
  const size_t planeBytes = planeHW * sizeof(float);
  const float* fbase = (lS == 0) ? f0 : (lS == 1) ? f1 : (lS == 2) ? f2 : f3;
  const char*  fpb   = (const char*)(fbase + (size_t)bS * 256 * planeHW
                                           + (size_t)cbS * 16 * planeHW);

  // Per-lane gather byte offsets: rows ys[4k+koff{,+1}], column xs[m]
  unsigned boff0[4], boff1[4];
  const int xcol = s_xs[m];
  #pragma unroll
  for (int k = 0; k < 4; ++k) {
    int r0 = 4 * k + koff;
    boff0[k] = (unsigned)(s_ys[r0]     * WS + xcol) * 4u;
    boff1[k] = (unsigned)(s_ys[r0 + 1] * WS + xcol) * 4u;
  }

  // A_y (A-operand, 16x4 per step) and A_x^T (B-operand, 4x16 per step)
  v2f ay[4], bx[4];
  #pragma unroll
  for (int k = 0; k < 4; ++k) {
    int c0 = 4 * k + koff;
    float a0 = 0.0f, a1 = 0.0f, b0 = 0.0f, b1 = 0.0f;
    if (m < 7) {
      if      (c0     == 2 * m)     a0 = s_wyh[m];
      else if (c0     == 2 * m + 1) a0 = s_wyl[m];
      if      (c0 + 1 == 2 * m)     a1 = s_wyh[m];
      else if (c0 + 1 == 2 * m + 1) a1 = s_wyl[m];
      if      (c0     == 2 * m)     b0 = s_wxh[m];
      else if (c0     == 2 * m + 1) b0 = s_wxl[m];
      if      (c0 + 1 == 2 * m)     b1 = s_wxh[m];
      else if (c0 + 1 == 2 * m + 1) b1 = s_wxl[m];
    }
    ay[k].x = a0; ay[k].y = a1;
    bx[k].x = b0; bx[k].y = b1;
  }

  float*    Ts   = &s_T[wv][0][0];
  const int trow = (lane < 16) ? 0 : 8;          // C/D lane-half M split

  // Scalar output base for this wave's channel slab; lane offset is 32-bit.
  float* outw = out + ((size_t)box * 256 + (size_t)cbS * 16) * 49;

  for (int ci = 0; ci < 16; ++ci) {
    const char* fpc = fpb + (size_t)ci * planeBytes;   // scalar plane base

    // Gather F (14x14 corner grid, padded 16x16) straight into B layout.
    // The empty asm pins the offsets as 32-bit VGPRs inside the loop so the
    // backend keeps saddr + zext(v32) and selects SADDR-form loads.
    v2f bm[4];
    #pragma unroll
    for (int k = 0; k < 4; ++k) {
      unsigned o0 = boff0[k], o1 = boff1[k];
      asm("" : "+v"(o0), "+v"(o1));
      bm[k].x = *(const float*)(fpc + o0);
      bm[k].y = *(const float*)(fpc + o1);
    }
    if (ci + 1 < 16) {
      // WGP-scope prefetch (default scope) of next channel plane rows.
      const char* pf0 = fpc + planeBytes + boff0[0];
      const char* pf1 = fpc + planeBytes + boff1[2];
      asm volatile("global_prefetch_b8 %0, off" :: "v"(pf0));
      asm volatile("global_prefetch_b8 %0, off" :: "v"(pf1));
    }

    // GEMM1: T = A_y * F   (y interpolation)
    v8f t = {0.f, 0.f, 0.f, 0.f, 0.f, 0.f, 0.f, 0.f};
    #pragma unroll
    for (int k = 0; k < 4; ++k)
      t = __builtin_amdgcn_wmma_f32_16x16x4_f32(false, ay[k], false, bm[k],
                                                (short)0, t, false, false);

    // C-layout -> A-layout via LDS (DS ops are in-order per wave)
    #pragma unroll
    for (int v = 0; v < 8; ++v)
      Ts[(v + trow) * 17 + m] = t[v];
    asm volatile("s_wait_dscnt 0" ::: "memory");

    v2f a2[4];
    #pragma unroll
    for (int k = 0; k < 4; ++k) {
      int c0 = 4 * k + koff;
      a2[k].x = Ts[m * 17 + c0];
      a2[k].y = Ts[m * 17 + c0 + 1];
    }

    // GEMM2: O = T * A_x^T   (x interpolation)
    v8f o = {0.f, 0.f, 0.f, 0.f, 0.f, 0.f, 0.f, 0.f};
    #pragma unroll
    for (int k = 0; k < 4; ++k)
      o = __builtin_amdgcn_wmma_f32_16x16x4_f32(false, a2[k], false, bx[k],
                                                (short)0, o, false, false);

    // D rows 0..6 live in VGPRs 0..6 of lanes 0..15; lane = pw.
    // Output is write-once -> non-temporal stores keep the pyramid in L2.
    if (lane < 7) {
      unsigned oo = (unsigned)(ci * 49 + lane);
      asm("" : "+v"(oo));
      float* op = outw + oo;
      #pragma unroll
      for (int v = 0; v < 7; ++v)
        __builtin_nontemporal_store(o[v], op + v * 7);
    }
  }
}

extern "C" void kernel_launch(void* const* d_in, const int* in_sizes, int n_in,
                              void* d_out, int out_size, void* d_ws, size_t ws_size,
                              hipStream_t stream) {
  const float* f0 = (const float*)d_in[0];
  const float* f1 = (const float*)d_in[1];
  const float* f2 = (const float*)d_in[2];
  const float* f3 = (const float*)d_in[3];
  const float* b0 = (const float*)d_in[4];
  const float* b1 = (const float*)d_in[5];
  int n0   = in_sizes[4] / 4;
  int n1   = in_sizes[5] / 4;
  int ntot = n0 + n1;

  dim3 grid((unsigned)ntot, 4);
  msroi_wmma_kernel<<<grid, 128, 0, stream>>>(f0, f1, f2, f3, b0, b1,
                                              (float*)d_out, n0);
  (void)n_in; (void)out_size; (void)d_ws; (void)ws_size;
}